// dominant_modal_estimator_51969104281908
// MI455X (gfx1250) — compile-verified
//
#include <hip/hip_runtime.h>

// Problem geometry (fixed by the reference): x is (N=2, D=192, H=256, W=512) f32.
// out is (N,1,H,W) f32 -> 262144 elements.
#define BS   64            // threads per block = pixels per block
#define DD   192           // disparity depth
#define HW_  131072        // H*W = 256*512 (stride between d-slices, in floats)

// ---- CDNA5 async global->LDS path (guarded so compile never breaks) ----------
#if defined(__has_builtin)
#  if __has_builtin(__builtin_amdgcn_global_load_async_to_lds_b32)
#    define HAVE_ASYNC_LDS 1
#  else
#    define HAVE_ASYNC_LDS 0
#  endif
#  if __has_builtin(__builtin_amdgcn_s_wait_asynccnt)
#    define HAVE_WAIT_ASYNC 1
#  else
#    define HAVE_WAIT_ASYNC 0
#  endif
#else
#  define HAVE_ASYNC_LDS 0
#  define HAVE_WAIT_ASYNC 0
#endif

#if HAVE_ASYNC_LDS
typedef __attribute__((address_space(1))) int GI;    // global int (builtin's param type)
typedef __attribute__((address_space(3))) int LI;    // LDS int
#endif

// Blur recompute at a single d, add order identical to the streaming window
// (reference: acc = ((((x[d-2]+x[d-1])+x[d])+x[d+1])+x[d+2]) / 5).
// Skipping an out-of-range term vs adding +0.0f is bit-exact for x >= 0.
__device__ __forceinline__ float blur_at(const float* xl, int d) {
    float s = 0.0f;
    if (d >= 2)     s = s + xl[(d - 2) * BS];
    if (d >= 1)     s = s + xl[(d - 1) * BS];
    s = s + xl[d * BS];
    if (d + 1 < DD) s = s + xl[(d + 1) * BS];
    if (d + 2 < DD) s = s + xl[(d + 2) * BS];
    return s / 5.0f;
}

// Blurred value with the first-mode interval [zlo, zhi] zeroed (x_blur2).
__device__ __forceinline__ float b2_at(const float* xl, int d, int zlo, int zhi) {
    if (d >= zlo && d <= zhi) return 0.0f;
    return blur_at(xl, d);
}

// index_r: first s in (index, D] with diff(prepend/append ones)[s] > 0, minus 1.
// If none (argmax over all-zero ind_r == 0) -> -1.
__device__ __forceinline__ int scan_right(const float* xl, int index, int zlo, int zhi) {
    float bprev = b2_at(xl, index, zlo, zhi);
    for (int s = index + 1; s <= DD; ++s) {
        float bcur = (s < DD) ? b2_at(xl, s, zlo, zhi) : 1.0f;  // append ones
        if (bcur - bprev > 0.0f) return s - 1;
        bprev = bcur;
    }
    return -1;
}

// index_l: last s <= index with diff(prepend ones)[s] < 0.
// If none (argmax over flipped all-zero ind_l == 0) -> D-1.
__device__ __forceinline__ int scan_left(const float* xl, int index, int zlo, int zhi) {
    float bcur = b2_at(xl, index, zlo, zhi);
    for (int s = index; s >= 1; --s) {
        float bprev = b2_at(xl, s - 1, zlo, zhi);
        if (bcur - bprev < 0.0f) return s;
        bcur = bprev;
    }
    // s == 0: dl[0] = b[0] - 1
    return (bcur - 1.0f < 0.0f) ? 0 : (DD - 1);
}

// _modal_mask on blur with interval [zlo,zhi] zeroed. The mask is always an
// interval [lo,hi] (possibly empty: lo > hi).
__device__ __forceinline__ void modal_interval(const float* xl, int zlo, int zhi,
                                               int& lo, int& hi) {
    // streaming argmax of the (zeroed) blurred vector; 1 LDS read per step
    float xm2 = 0.0f, xm1 = 0.0f;
    float x0  = xl[0];
    float xp1 = xl[1 * BS];
    float xp2 = xl[2 * BS];
    float bmax = -3.402823466e+38f;
    int   idx  = 0;
    for (int d = 0; d < DD; ++d) {
        float b = ((((xm2 + xm1) + x0) + xp1) + xp2) / 5.0f;
        if (d >= zlo && d <= zhi) b = 0.0f;
        if (b > bmax) { bmax = b; idx = d; }         // strict > : first occurrence
        xm2 = xm1; xm1 = x0; x0 = xp1; xp1 = xp2;
        xp2 = (d + 3 < DD) ? xl[(d + 3) * BS] : 0.0f;
    }
    int ir = scan_right(xl, idx, zlo, zhi);
    int il = scan_left (xl, idx, zlo, zhi);
    int dev = 2 * idx - ir - il;
    bool valid = (dev < 3) && (dev > -3);
    if (valid) {
        lo = il; hi = ir;
    } else {
        int ra = ir - idx, rb = idx - il;
        int r  = ra < rb ? ra : rb;                  // may be negative -> empty
        lo = idx - r; hi = idx + r;
    }
}

extern "C" __global__ __launch_bounds__(BS)
void dme_kernel(const float* __restrict__ x, float* __restrict__ out, int npix) {
    __shared__ float xs[DD * BS];                    // 48 KB, bank = tid (conflict-free)
    const int tid = threadIdx.x;
    const int p   = blockIdx.x * BS + tid;
    if (p >= npix) return;

    const int n = p / HW_;                           // batch index
    const float* gbase = x + (size_t)p + (size_t)n * (size_t)(DD - 1) * (size_t)HW_;
    float* xl = &xs[tid];                            // column base, stride BS

    // ---- load this thread's 192-deep column into LDS -------------------------
#if HAVE_ASYNC_LDS
    #pragma unroll 8
    for (int d = 0; d < DD; ++d) {
        __builtin_amdgcn_global_load_async_to_lds_b32(
            (GI*)(gbase + (size_t)d * HW_),          // global src (as1 int*)
            (LI*)(xl + d * BS),                      // LDS dst    (as3 int*)
            0, 0);                                   // imm offset, cpol
    }
#if HAVE_WAIT_ASYNC
    __builtin_amdgcn_s_wait_asynccnt(0);
#else
    asm volatile("s_wait_asynccnt 0" ::: "memory");
#endif
#else
    for (int d = 0; d < DD; ++d)
        xl[d * BS] = gbase[(size_t)d * HW_];
#endif
    // Each thread reads only the LDS words it wrote itself: no barrier needed.

    // ---- first modal mask (on blur), second (on blur with mode 1 zeroed) -----
    int lo1, hi1, lo2, hi2;
    modal_interval(xl, 0, -1, lo1, hi1);             // no zero interval
    modal_interval(xl, lo1, hi1, lo2, hi2);          // x_blur2 = x_blur * ~mask

    // ---- accumulate y (mode 1) and z (mode 2 minus mode 1) -------------------
    float sum_y = 0.0f, wsum_y = 0.0f, sum_z = 0.0f, wsum_z = 0.0f;
    int dmin = lo1 < lo2 ? lo1 : lo2;
    int dmax = hi1 > hi2 ? hi1 : hi2;
    if (dmin < 0)      dmin = 0;
    if (dmax > DD - 1) dmax = DD - 1;
    for (int d = dmin; d <= dmax; ++d) {
        float xv = xl[d * BS];
        float fd = (float)d;
        bool in1 = (d >= lo1) && (d <= hi1);
        bool in2 = (d >= lo2) && (d <= hi2);
        if (in1)      { sum_y += xv; wsum_y += xv * fd; }
        else if (in2) { sum_z += xv; wsum_z += xv * fd; }
    }

    // valid = sum(y) >= sum(z); out = sum(sel*d)/sum(sel)
    float res = (sum_y >= sum_z) ? (wsum_y / sum_y) : (wsum_z / sum_z);
    out[p] = res;
}

extern "C" void kernel_launch(void* const* d_in, const int* in_sizes, int n_in,
                              void* d_out, int out_size, void* d_ws, size_t ws_size,
                              hipStream_t stream) {
    (void)in_sizes; (void)n_in; (void)d_ws; (void)ws_size;
    const float* x = (const float*)d_in[0];
    float* out = (float*)d_out;
    int npix = out_size;                             // N*H*W = 262144
    dim3 block(BS);
    dim3 grid((npix + BS - 1) / BS);
    hipLaunchKernelGGL(dme_kernel, grid, block, 0, stream, x, out, npix);
}